// TokenAwareEmbedding_78323023610034
// MI455X (gfx1250) — compile-verified
//
#include <hip/hip_runtime.h>

// Token-aware NF4 embedding gather for gfx1250 (MI455X).
// Memory-bound: ~134 MB total traffic -> ~6us at 23.3 TB/s HBM.
// CDNA5 paths used:
//  - global_load_async_to_lds_b128 (ASYNCcnt) double-buffers each token's
//    4KB quantized row per wave32 (wave-private LDS -> no loop barriers)
//  - s_wait_asynccnt split-counter waits
//  - non-temporal output stores (write-once 67MB) so the 192MB L2 keeps the
//    206MB NF4 table / scales / special rows, which have gather reuse.

#define VOCAB  50257
#define DIM    1024
#define NTOK   (8 * 2048)
#define NSPEC  256
#define WAVES_PER_BLK 8
#define GRID_BLOCKS   512   // 4096 waves -> 4 tokens per wave (pipeline depth)

typedef _Float16 half4_t __attribute__((ext_vector_type(4)));
typedef float    f32x4   __attribute__((ext_vector_type(4)));
typedef int      i32x4   __attribute__((ext_vector_type(4)));

__device__ __constant__ float kNF4[16] = {
    -1.0f, -0.6962f, -0.5251f, -0.3949f, -0.2844f, -0.1848f, -0.0911f, 0.0f,
     0.0796f, 0.1609f, 0.2461f, 0.3379f, 0.4407f, 0.5626f, 0.723f, 1.0f};

__global__ __launch_bounds__(256) void nf4_embed_gather(
    const int* __restrict__ ids,          // [NTOK]
    const int* __restrict__ qcodes,       // [VOCAB*DIM] int32 codes 0..15
    const float* __restrict__ scales,     // [VOCAB*16]
    const _Float16* __restrict__ spec,    // [NSPEC*DIM] fp16
    const int* __restrict__ specIdx,      // [NSPEC] sorted
    float* __restrict__ out)              // [NTOK*DIM] fp32
{
    __shared__ float sLUT[16];
    __shared__ int   sSpec[NSPEC];
    __shared__ alignas(16) int sCodes[WAVES_PER_BLK][2][DIM];  // 64 KB double buffer

    const int tid = threadIdx.x;
    if (tid < 16)    sLUT[tid]  = kNF4[tid];
    if (tid < NSPEC) sSpec[tid] = specIdx[tid];
    __syncthreads();

    const int lane = tid & 31;
    const int wIn  = tid >> 5;
    const int wave = blockIdx.x * WAVES_PER_BLK + wIn;
    const int nWaves = gridDim.x * WAVES_PER_BLK;

    const unsigned ldsBase0 = (unsigned)(uintptr_t)&sCodes[wIn][0][0];
    const unsigned ldsBase1 = (unsigned)(uintptr_t)&sCodes[wIn][1][0];
    const unsigned laneOff  = (unsigned)(lane * 4) * 4u;  // 4 codes * 4B per lane

    int t = wave;
    if (t >= NTOK) return;

    // Prologue: async-stage first token's 4KB code row into buffer 0.
    {
        const int v = ids[t];
        const unsigned long long g0 =
            (unsigned long long)(uintptr_t)qcodes + (unsigned long long)v * (DIM * 4ull);
#pragma unroll
        for (int k = 0; k < 8; ++k) {
            const unsigned off = (unsigned)(k * 128) * 4u + laneOff;
            asm volatile("global_load_async_to_lds_b128 %0, %1, off"
                         :: "v"(ldsBase0 + off), "v"(g0 + off) : "memory");
        }
    }

    int buf = 0;
    while (t < NTOK) {
        const int v  = ids[t];
        const int tn = t + nWaves;

        if (tn < NTOK) {
            // Prefetch next token's row into the other buffer, then wait until
            // only those 8 new async ops remain -> current buffer is complete
            // (async loads complete in order).
            const int vn = ids[tn];
            const unsigned ldsN = (buf ^ 1) ? ldsBase1 : ldsBase0;
            const unsigned long long g0 =
                (unsigned long long)(uintptr_t)qcodes + (unsigned long long)vn * (DIM * 4ull);
#pragma unroll
            for (int k = 0; k < 8; ++k) {
                const unsigned off = (unsigned)(k * 128) * 4u + laneOff;
                asm volatile("global_load_async_to_lds_b128 %0, %1, off"
                             :: "v"(ldsN + off), "v"(g0 + off) : "memory");
            }
            asm volatile("s_wait_asynccnt 8" ::: "memory");
        } else {
            asm volatile("s_wait_asynccnt 0" ::: "memory");
        }

        // Special-token lookup: upper_bound over sorted indices; with duplicate
        // indices the LAST occurrence wins (matches in-order scatter .set).
        int lo = 0, hi = NSPEC;
#pragma unroll
        for (int it = 0; it < 8; ++it) {  // log2(256) bisection steps
            const int m = (lo + hi) >> 1;
            if (sSpec[m] <= v) lo = m + 1; else hi = m;
        }
        const int r = (lo > 0 && sSpec[lo - 1] == v) ? (lo - 1) : -1;

        float* op = out + (size_t)t * DIM;

        if (r >= 0) {
            // Override row: fp16 special embedding -> fp32
            const _Float16* sp = spec + (size_t)r * DIM;
#pragma unroll
            for (int k = 0; k < 8; ++k) {
                const int d = k * 128 + lane * 4;
                const half4_t h = *(const half4_t*)(sp + d);
                f32x4 o;
                o.x = (float)h.x; o.y = (float)h.y; o.z = (float)h.z; o.w = (float)h.w;
                __builtin_nontemporal_store(o, (f32x4*)(op + d));
            }
        } else {
            // NF4 dequant from LDS-staged codes
            const int* cp = buf ? &sCodes[wIn][1][0] : &sCodes[wIn][0][0];
            const int sbase = v * (DIM / 64);  // 16 scale blocks per row
#pragma unroll
            for (int k = 0; k < 8; ++k) {
                const int d = k * 128 + lane * 4;
                const i32x4 c = *(const i32x4*)(cp + d);
                const float s = scales[sbase + (d >> 6)];
                f32x4 o;
                o.x = sLUT[c.x & 15] * s;
                o.y = sLUT[c.y & 15] * s;
                o.z = sLUT[c.z & 15] * s;
                o.w = sLUT[c.w & 15] * s;
                __builtin_nontemporal_store(o, (f32x4*)(op + d));
            }
        }

        t = tn;
        buf ^= 1;
    }
}

extern "C" void kernel_launch(void* const* d_in, const int* in_sizes, int n_in,
                              void* d_out, int out_size, void* d_ws, size_t ws_size,
                              hipStream_t stream) {
    (void)in_sizes; (void)n_in; (void)d_ws; (void)ws_size; (void)out_size;
    const int*      ids = (const int*)d_in[0];       // input_ids (int32; jax x64 off)
    const int*      q   = (const int*)d_in[1];       // main_quantized
    const float*    sc  = (const float*)d_in[2];     // main_scales
    const _Float16* sp  = (const _Float16*)d_in[3];  // special_embeddings fp16
    const int*      si  = (const int*)d_in[4];       // special_indices (sorted)
    float*          out = (float*)d_out;

    nf4_embed_gather<<<GRID_BLOCKS, 256, 0, stream>>>(ids, q, sc, sp, si, out);
}